// GATLayer_13091060318781
// MI455X (gfx1250) — compile-verified
//
#include <hip/hip_runtime.h>
#include <hip/hip_bf16.h>
#include <math.h>

#define N_NODES   100000
#define N_EDGES   1600000
#define IN_DIM    128
#define OUT_DIM   128
#define N_HEADS   4
#define HEAD_DIM  32
#define NEG_SLOPE 0.2f

#define MROWS 32   // rows of h staged per block (2 x 16-row WMMA tiles)

typedef __attribute__((ext_vector_type(2))) float        v2f;
typedef __attribute__((ext_vector_type(8))) float        v8f;
typedef __attribute__((ext_vector_type(4))) unsigned int v4u;
typedef __attribute__((ext_vector_type(8))) int          v8i;
typedef __attribute__((ext_vector_type(4))) int          v4i;

// ---------------------------------------------------------------------------
// TDM: DMA one 32x128 f32 tile of h into LDS (offset 0) with 4-DWORD row
// padding (rows land at 132-float pitch). D# per CDNA5 ISA:
//   group0: count=1 | lds_addr=0 | global_addr | type=2
//   group1: data_size=4B(code2), pad_enable, pad_interval=128dw(code6),
//           pad_amount=4dw(code3), tensor 128x32, tile 128x32, stride0=128
// ---------------------------------------------------------------------------
#if __has_builtin(__builtin_amdgcn_tensor_load_to_lds)
#define HAVE_TDM 1
__device__ __forceinline__ void tdm_load_tile_32x128(const float* gptr) {
    const unsigned long long ga = (unsigned long long)(uintptr_t)gptr;
    v4u g0;
    g0[0] = 1u;                                        // count=1 (valid user D#)
    g0[1] = 0u;                                        // lds_addr = 0 (As @ LDS 0)
    g0[2] = (unsigned)(ga & 0xffffffffu);              // global_addr[31:0]
    g0[3] = (unsigned)((ga >> 32) & 0x1ffffffu)        // global_addr[56:32]
          | (2u << 30);                                // type = 2 ("image")
    v8i g1;
    g1[0] = (int)((2u << 16)                           // data_size = 4 bytes
                | (1u << 20)                           // pad_enable
                | (6u << 22)                           // pad_interval = 128 DWORDs
                | (3u << 25));                         // pad_amount  = 4 DWORDs
    g1[1] = (int)(128u << 16);                         // tensor_dim0 = 128 (lo16)
    g1[2] = (int)((unsigned)MROWS << 16);              // dim0 hi=0 | tensor_dim1 = 32
    g1[3] = (int)(128u << 16);                         // dim1 hi=0 | tile_dim0 = 128
    g1[4] = MROWS;                                     // tile_dim1 = 32, tile_dim2 = 0
    g1[5] = 128;                                       // tensor_dim0_stride = 128
    g1[6] = 0;
    g1[7] = 0;
    v4i gz = {0, 0, 0, 0};
#if __clang_major__ >= 23
    v8i gz8 = {0, 0, 0, 0, 0, 0, 0, 0};
    __builtin_amdgcn_tensor_load_to_lds(g0, g1, gz, gz, gz8, 0);
#else
    __builtin_amdgcn_tensor_load_to_lds(g0, g1, gz, gz, 0);
#endif
}
#else
#define HAVE_TDM 0
#endif

// ---------------------------------------------------------------------------
// Kernel 1: wh = h @ W using V_WMMA_F32_16X16X4_F32 (f32 in, f32 accum).
// 256 threads = 8 waves. Block stages a 32x128 h tile (TDM) in LDS; wave w
// owns 16-column tile w and computes TWO 16x16 output tiles with a shared B
// fragment (2x B reuse, interleaved WMMAs hide the D->C hazard). 64 WMMAs.
// ---------------------------------------------------------------------------
__global__ void __launch_bounds__(256)
gat_gemm_wh(const float* __restrict__ h, const float* __restrict__ W,
            float* __restrict__ wh) {
    __shared__ float As[MROWS * 132];           // MUST be sole LDS alloc (offset 0)
    const int rowBase = blockIdx.x * MROWS;
    const int t = threadIdx.x;

#if HAVE_TDM
    if (t < 32) {                               // wave 0 issues the DMA
        tdm_load_tile_32x128(h + (size_t)rowBase * IN_DIM);
        __builtin_amdgcn_s_wait_tensorcnt(0);
    }
    // The D# aliases As but the compiler can't see it: force it to assume
    // LDS was written (keeps the ds_loads live).
    asm volatile("" : : "r"(&As[0]) : "memory");
#else
    for (int i = t; i < MROWS * 128; i += 256) {
        const int r = i >> 7, c = i & 127;
        As[r * 132 + c] = h[(rowBase + r) * IN_DIM + c];
    }
#endif
    __syncthreads();

    const int wave = t >> 5;                    // 0..7 -> column tile
    const int lane = t & 31;
    const int colBase = wave * 16;
    const int lrow = lane & 15;
    const int khalf = (lane >> 4) << 1;         // 0 for lanes 0-15, 2 for 16-31

    v8f acc0 = {}, acc1 = {};
    #pragma unroll
    for (int k = 0; k < IN_DIM; k += 4) {
        v2f b, a0, a1;
        b[0]  = W[(k + khalf) * OUT_DIM + colBase + lrow];
        b[1]  = W[(k + khalf + 1) * OUT_DIM + colBase + lrow];
        a0[0] = As[lrow * 132 + k + khalf];
        a0[1] = As[lrow * 132 + k + khalf + 1];
        a1[0] = As[(16 + lrow) * 132 + k + khalf];
        a1[1] = As[(16 + lrow) * 132 + k + khalf + 1];
        acc0 = __builtin_amdgcn_wmma_f32_16x16x4_f32(
            false, a0, false, b, (short)0, acc0, false, false);
        acc1 = __builtin_amdgcn_wmma_f32_16x16x4_f32(
            false, a1, false, b, (short)0, acc1, false, false);
    }

    const int mOff = (lane >> 4) ? 8 : 0;
    #pragma unroll
    for (int r = 0; r < 8; ++r) {
        wh[(rowBase + mOff + r) * OUT_DIM + colBase + lrow]      = acc0[r];
        wh[(rowBase + 16 + mOff + r) * OUT_DIM + colBase + lrow] = acc1[r];
    }
}

// ---------------------------------------------------------------------------
// Kernel 2: one wave per node; lane l covers dims [4l,4l+4) (float4 coalesced),
// 8-lane shuffle reduction per head.
// ---------------------------------------------------------------------------
__global__ void gat_scores(const float* __restrict__ wh, const float* __restrict__ a,
                           float* __restrict__ s_src, float* __restrict__ s_dst) {
    const int node = (blockIdx.x * blockDim.x + threadIdx.x) >> 5;
    const int lane = threadIdx.x & 31;
    if (node >= N_NODES) return;
    const int head = lane >> 3;                 // 8 lanes per head
    const int sub  = lane & 7;
    const float4 v  = *(const float4*)(wh + node * OUT_DIM + lane * 4);
    const float4 as = *(const float4*)(a + head * (2 * HEAD_DIM) + sub * 4);
    const float4 ad = *(const float4*)(a + head * (2 * HEAD_DIM) + HEAD_DIM + sub * 4);
    float ss = v.x * as.x + v.y * as.y + v.z * as.z + v.w * as.w;
    float sd = v.x * ad.x + v.y * ad.y + v.z * ad.z + v.w * ad.w;
    #pragma unroll
    for (int off = 4; off > 0; off >>= 1) {
        ss += __shfl_down(ss, off, 8);
        sd += __shfl_down(sd, off, 8);
    }
    if (sub == 0) {
        s_src[node * N_HEADS + head] = ss;
        s_dst[node * N_HEADS + head] = sd;
    }
}

// ---------------------------------------------------------------------------
// Kernel 3: init out = 0, m = -inf, denom = 0
// ---------------------------------------------------------------------------
__global__ void gat_init(float* __restrict__ out, float* __restrict__ m,
                         float* __restrict__ denom) {
    const int gid = blockIdx.x * blockDim.x + threadIdx.x;
    if (gid < N_NODES * OUT_DIM) out[gid] = 0.f;
    if (gid < N_NODES * N_HEADS) {
        m[gid] = __int_as_float(0xff800000);    // -inf
        denom[gid] = 0.f;
    }
}

// Float atomic max via signed/unsigned integer punning (monotone bit trick)
__device__ __forceinline__ void atomicMaxFloat(float* addr, float val) {
    if (val >= 0.0f) atomicMax((int*)addr, __float_as_int(val));
    else             atomicMin((unsigned int*)addr, __float_as_uint(val));
}

// ---------------------------------------------------------------------------
// Kernel 4: one thread per edge, all 4 heads via float4; segment max into m.
// ---------------------------------------------------------------------------
__global__ void gat_edge_logits(const int* __restrict__ src, const int* __restrict__ dst,
                                const float* __restrict__ s_src, const float* __restrict__ s_dst,
                                float* __restrict__ e, float* __restrict__ m) {
    const int eid = blockIdx.x * blockDim.x + threadIdx.x;
    if (eid >= N_EDGES) return;
    const int s = src[eid], d = dst[eid];
    const float4 a4 = *(const float4*)(s_src + s * N_HEADS);
    const float4 b4 = *(const float4*)(s_dst + d * N_HEADS);
    float4 x;
    x.x = a4.x + b4.x; x.y = a4.y + b4.y; x.z = a4.z + b4.z; x.w = a4.w + b4.w;
    x.x = (x.x > 0.f) ? x.x : NEG_SLOPE * x.x;
    x.y = (x.y > 0.f) ? x.y : NEG_SLOPE * x.y;
    x.z = (x.z > 0.f) ? x.z : NEG_SLOPE * x.z;
    x.w = (x.w > 0.f) ? x.w : NEG_SLOPE * x.w;
    *(float4*)(e + eid * N_HEADS) = x;
    float* mp = m + d * N_HEADS;
    atomicMaxFloat(mp + 0, x.x);
    atomicMaxFloat(mp + 1, x.y);
    atomicMaxFloat(mp + 2, x.z);
    atomicMaxFloat(mp + 3, x.w);
}

// ---------------------------------------------------------------------------
// Kernel 5: ex = exp(e - m[dst]); denom[dst] += ex (in-place overwrite of e)
// ---------------------------------------------------------------------------
__global__ void gat_edge_exp(const int* __restrict__ dst, float* __restrict__ e,
                             const float* __restrict__ m, float* __restrict__ denom) {
    const int eid = blockIdx.x * blockDim.x + threadIdx.x;
    if (eid >= N_EDGES) return;
    const int d = dst[eid];
    const float4 x  = *(const float4*)(e + eid * N_HEADS);
    const float4 mm = *(const float4*)(m + d * N_HEADS);
    float4 ex;
    ex.x = expf(x.x - mm.x);
    ex.y = expf(x.y - mm.y);
    ex.z = expf(x.z - mm.z);
    ex.w = expf(x.w - mm.w);
    *(float4*)(e + eid * N_HEADS) = ex;
    float* dp = denom + d * N_HEADS;
    atomicAdd(dp + 0, ex.x);
    atomicAdd(dp + 1, ex.y);
    atomicAdd(dp + 2, ex.z);
    atomicAdd(dp + 3, ex.w);
}

// ---------------------------------------------------------------------------
// Kernel 6: out[dst] += (ex/denom[dst]) * wh[src].
// One wave per edge: lane l handles dims [4l,4l+4) -> global_load_b128
// gathers of wh[src] and 4 coalesced f32 atomic adds per lane.
// ---------------------------------------------------------------------------
__global__ void gat_aggregate(const int* __restrict__ src, const int* __restrict__ dst,
                              const float* __restrict__ wh, const float* __restrict__ ex,
                              const float* __restrict__ denom, float* __restrict__ out) {
    const int gid = blockIdx.x * blockDim.x + threadIdx.x;   // edge*32 + lane
    if (gid >= N_EDGES * 32) return;
    const int eid  = gid >> 5;
    const int lane = gid & 31;
    const int head = lane >> 3;
    const int s = src[eid], d = dst[eid];
    const float w = ex[eid * N_HEADS + head] / denom[d * N_HEADS + head];
    const float4 v = *(const float4*)(wh + s * OUT_DIM + lane * 4);
    float* o = out + d * OUT_DIM + lane * 4;
    atomicAdd(o + 0, w * v.x);
    atomicAdd(o + 1, w * v.y);
    atomicAdd(o + 2, w * v.z);
    atomicAdd(o + 3, w * v.w);
}

// ---------------------------------------------------------------------------
// Launch
// ---------------------------------------------------------------------------
extern "C" void kernel_launch(void* const* d_in, const int* in_sizes, int n_in,
                              void* d_out, int out_size, void* d_ws, size_t ws_size,
                              hipStream_t stream) {
    const float* h  = (const float*)d_in[0];
    const int*   ei = (const int*)d_in[1];          // [2, N_EDGES]
    const float* W  = (const float*)d_in[2];
    const float* a  = (const float*)d_in[3];
    float* out = (float*)d_out;

    const int* src = ei;
    const int* dst = ei + N_EDGES;

    // Workspace layout (bytes, all 16B-aligned)
    char* ws = (char*)d_ws;
    float* wh    = (float*)(ws + 0);                // 51.2 MB
    float* s_src = (float*)(ws + 51200000);         // 1.6 MB
    float* s_dst = (float*)(ws + 52800000);         // 1.6 MB
    float* e     = (float*)(ws + 54400000);         // 25.6 MB
    float* m     = (float*)(ws + 80000000);         // 1.6 MB
    float* denom = (float*)(ws + 81600000);         // 1.6 MB

    // 1) GEMM (100000 = 3125 * 32 row tiles)
    gat_gemm_wh<<<N_NODES / MROWS, 256, 0, stream>>>(h, W, wh);

    // 2) attention scores: one wave per node
    gat_scores<<<(N_NODES * 32 + 255) / 256, 256, 0, stream>>>(wh, a, s_src, s_dst);

    // 3) init out / m / denom
    gat_init<<<(N_NODES * OUT_DIM + 255) / 256, 256, 0, stream>>>(out, m, denom);

    // 4) edge logits + segment max
    gat_edge_logits<<<(N_EDGES + 255) / 256, 256, 0, stream>>>(src, dst, s_src, s_dst, e, m);

    // 5) exp + segment sum
    gat_edge_exp<<<(N_EDGES + 255) / 256, 256, 0, stream>>>(dst, e, m, denom);

    // 6) weighted aggregation: one wave per edge
    gat_aggregate<<<(N_EDGES * 32 + 255) / 256, 256, 0, stream>>>(src, dst, wh, e, denom, out);
}